// LossFunction_62852551409895
// MI455X (gfx1250) — compile-verified
//
#include <hip/hip_runtime.h>
#include <math.h>

// ---------------- problem constants (from reference) ----------------
#define BB   32768            // batch
#define CC   1000             // classes
#define KK   2                // aux heads
#define ROWS ((KK + 1) * BB)  // 98304 softmax rows total
#define RPB  16               // rows per block (one LDS tile)
#define NBLK (ROWS / RPB)     // 6144 blocks, exact (B % 16 == 0)
#define STRIDE 1004           // LDS row stride in floats: 16B-aligned, bank-friendly
#define ALPHA 0.1f

typedef float v2f __attribute__((ext_vector_type(2)));
typedef float v8f __attribute__((ext_vector_type(8)));

// ---- CDNA5 async global->LDS path (guarded; fallback = sync b128 copy) ----
#if defined(__AMDGCN__) && defined(__has_builtin)
# if __has_builtin(__builtin_amdgcn_global_load_async_to_lds_b128)
#  define HAVE_ASYNC_LDS 1
# endif
#endif
#ifndef HAVE_ASYNC_LDS
# define HAVE_ASYNC_LDS 0
#endif

__global__ __launch_bounds__(256) void loss_rows_kernel(
    const float* __restrict__ outputs,   // [B, C]
    const float* __restrict__ aux,       // [K, B, C] flat
    const int*   __restrict__ labels,    // [B]
    const float* __restrict__ wb,        // [3]
    const float* __restrict__ abias,     // [1]
    const float* __restrict__ agamma,    // [1]
    float*       __restrict__ blocksum)  // [NBLK]
{
  __shared__ __align__(16) float tile[RPB * STRIDE];   // 64,256 B
  __shared__ float sv0[RPB], sv1[RPB], sx[RPB];
  __shared__ float part[8][RPB];

  const int tid  = threadIdx.x;
  const int lane = tid & 31;
  const int w    = tid >> 5;            // wave id 0..7 (wave-uniform)
  const int bid  = blockIdx.x;
  const int r0   = bid * RPB;           // first global row of this tile

  // -------- stage 1: stream 16 rows x 1000 f32 into LDS (single HBM read) --------
  for (int idx4 = tid; idx4 < RPB * (CC / 4); idx4 += 256) {
    int row  = idx4 / (CC / 4);
    int col4 = idx4 - row * (CC / 4);
    int r    = r0 + row;
    const float* src = (r < BB) ? (outputs + (size_t)r * CC)
                                : (aux + (size_t)(r - BB) * CC);
    src += col4 * 4;
    float* dst = &tile[row * STRIDE + col4 * 4];
#if HAVE_ASYNC_LDS
    typedef int gv4i __attribute__((vector_size(16)));
    __builtin_amdgcn_global_load_async_to_lds_b128(
        (__attribute__((address_space(1))) gv4i*)src,
        (__attribute__((address_space(3))) gv4i*)dst,
        0, 0);
#else
    *(float4*)dst = *(const float4*)src;
#endif
  }
#if HAVE_ASYNC_LDS
# if __has_builtin(__builtin_amdgcn_s_wait_asynccnt)
  __builtin_amdgcn_s_wait_asynccnt(0);
# else
  asm volatile("s_wait_asynccnt 0" ::: "memory");
# endif
#endif
  __syncthreads();

  // -------- stage 2: per-row top-2 (v0 = max) + x = row[label], from LDS --------
  for (int rr = 0; rr < 2; ++rr) {
    int row = w * 2 + rr;
    const float* rp = &tile[row * STRIDE];
    float m1 = -INFINITY, m2 = -INFINITY;
    for (int c = lane; c < CC; c += 32) {
      float v = rp[c];
      if (v > m1)      { m2 = m1; m1 = v; }
      else if (v > m2) { m2 = v; }
    }
    // merge sorted pairs across the wave (fixed-order tree, deterministic)
    for (int off = 16; off > 0; off >>= 1) {
      float o1 = __shfl_xor(m1, off, 32);
      float o2 = __shfl_xor(m2, off, 32);
      float n1 = fmaxf(m1, o1);
      float n2 = fmaxf(fminf(m1, o1), fmaxf(m2, o2));
      m1 = n1; m2 = n2;
    }
    if (lane == 0) {
      int r   = r0 + row;
      int lab = labels[r & (BB - 1)];   // r % B (B is a power of two)
      sv0[row] = m1;
      sv1[row] = m2;
      sx[row]  = rp[lab];
    }
  }
  __syncthreads();

  // -------- stage 3: sum(exp(v - max)) per row via V_WMMA_F32_16X16X4_F32 --------
  // A(16x4) = exp tile: lane m -> row m cols {c0,c0+1}; lane 16+m -> row m cols {c0+2,c0+3}
  // B(4x16) = ones  =>  D[m,n] = rowsum(A[m,:]) replicated over n; C accumulates.
  // Column-group g = w + 8*i  ->  col = 4*w + 32*i + 2*h. Trip count forced into an
  // SGPR (readfirstlane) so the loop is uniform: EXEC stays all-1s at every WMMA.
  {
    const int m = lane & 15;
    const int h = lane >> 4;                 // 0: K={0,1}, 1: K={2,3}
    const float* rp = &tile[m * STRIDE];
    const float rmax = sv0[m];
    const int niter = __builtin_amdgcn_readfirstlane((CC / 4 - w + 7) >> 3); // 31 or 32
    int c0 = 4 * w + 2 * h;
    v8f acc = {0.f, 0.f, 0.f, 0.f, 0.f, 0.f, 0.f, 0.f};
    v2f ones; ones.x = 1.0f; ones.y = 1.0f;
    float2 v = *(const float2*)(rp + c0);    // software-pipelined ds_load_b64
    for (int i = 0; i < niter; ++i) {
      float2 cur = v;
      if (i + 1 < niter)                     // uniform branch (SGPR counter)
        v = *(const float2*)(rp + c0 + (i + 1) * 32);
      v2f a;
      a.x = __expf(cur.x - rmax);
      a.y = __expf(cur.y - rmax);
      acc = __builtin_amdgcn_wmma_f32_16x16x4_f32(
          false, a, false, ones, (short)0, acc, false, false);
    }
    // D layout: lanes 0-15 vgpr g -> row g; lanes 16-31 vgpr g -> row 8+g
    if ((lane & 15) == 0) {
#pragma unroll
      for (int g = 0; g < 8; ++g)
        part[w][h * 8 + g] = acc[g];
    }
  }
  __syncthreads();

  // -------- stage 4: per-row loss terms, fixed-order block reduction --------
  if (tid < 32) {
    float contrib = 0.0f;
    if (tid < RPB) {
      const int m = tid;
      float s = 0.0f;
#pragma unroll
      for (int ww = 0; ww < 8; ++ww) s += part[ww][m];
      float v0  = sv0[m];
      float lse = v0 + __logf(s);
      float x   = sx[m];
      contrib = (lse - x) * (1.0f / (float)BB);   // CE mean contribution (all heads)
      if (r0 < BB) {                               // main head: distance term
        float th1 = wb[0], th2 = wb[1], b = wb[2];
        float v1 = sv1[m];
        float y  = (v0 == x) ? v1 : ((v1 == x) ? v0 : (v0 + v1));
        float dist = (th1 * x + th2 * y + b - abias[0]) / sqrtf(th1 * th1 + th2 * th2);
        float per  = (dist >= 10.0f) ? -2.0f
                   : ((dist >= 0.0f) ? (-agamma[0] * dist) : -dist);
        contrib += ALPHA * per;
      }
    }
    for (int off = 16; off > 0; off >>= 1)
      contrib += __shfl_xor(contrib, off, 32);
    if (tid == 0) blocksum[bid] = contrib;
  }
}

// Deterministic final reduction: fixed per-thread order + fixed LDS tree.
__global__ __launch_bounds__(256) void reduce_final_kernel(
    const float* __restrict__ bs, float* __restrict__ out)
{
  __shared__ float red[256];
  float s = 0.0f;
  for (int i = threadIdx.x; i < NBLK; i += 256) s += bs[i];
  red[threadIdx.x] = s;
  __syncthreads();
  for (int st = 128; st > 0; st >>= 1) {
    if (threadIdx.x < st) red[threadIdx.x] += red[threadIdx.x + st];
    __syncthreads();
  }
  if (threadIdx.x == 0) out[0] = red[0];
}

extern "C" void kernel_launch(void* const* d_in, const int* in_sizes, int n_in,
                              void* d_out, int out_size, void* d_ws, size_t ws_size,
                              hipStream_t stream) {
  (void)in_sizes; (void)n_in; (void)out_size; (void)ws_size;
  const float* outputs = (const float*)d_in[0];
  const float* aux     = (const float*)d_in[1];
  const int*   labels  = (const int*)d_in[2];
  const float* wb      = (const float*)d_in[3];
  const float* abias   = (const float*)d_in[4];
  const float* agamma  = (const float*)d_in[5];
  float* bs = (float*)d_ws;   // NBLK floats of scratch

  loss_rows_kernel<<<NBLK, 256, 0, stream>>>(outputs, aux, labels, wb, abias, agamma, bs);
  reduce_final_kernel<<<1, 256, 0, stream>>>(bs, (float*)d_out);
}